// HashTableVoxelizedGaussianAdapterModule_50749333569908
// MI455X (gfx1250) — compile-verified
//
#include <hip/hip_runtime.h>
#include <hip/hip_bf16.h>

#define BS 256
#define GRID1 4096   // blocks for the gather/reduce pass (grid-stride)

typedef unsigned int u32x4 __attribute__((ext_vector_type(4)));
typedef unsigned int u32x8 __attribute__((ext_vector_type(8)));

__device__ __forceinline__ unsigned hash3(int c0, int c1, int c2,
                                          unsigned tmask, unsigned T) {
    unsigned h = (unsigned)c0 * 1u
               ^ (unsigned)c1 * 2654435761u
               ^ (unsigned)c2 * 805459861u;
    return tmask ? (h & tmask) : (h % T);
}

__device__ __forceinline__ float sigmoidf(float x) {
    return 1.0f / (1.0f + expf(-x));
}

// ---------------------------------------------------------------------------
// Pass 1: gather features (software-pipelined with global_prefetch_b8),
// optionally cache them to ws, and produce per-block partial sums (double)
// of rows 0..2 for the global mean/std.
// ---------------------------------------------------------------------------
template <bool CACHE>
__global__ void k_gather_reduce(const int*   __restrict__ coords,
                                const float* __restrict__ table,
                                float*       __restrict__ cache,
                                double*      __restrict__ partials,
                                int N, unsigned T, unsigned tmask) {
    const int NF = CACHE ? 14 : 3;
    int i      = blockIdx.x * BS + threadIdx.x;
    int stride = gridDim.x * BS;

    double s = 0.0, ss = 0.0;

    unsigned idx = 0;
    if (i < N)
        idx = hash3(coords[3 * i], coords[3 * i + 1], coords[3 * i + 2], tmask, T);

    for (int cur = i; cur < N; cur += stride) {
        // Compute next iteration's index and prefetch its gather lines
        // (global_prefetch_b8: no LOADcnt, fills start while we load current).
        int nxt = cur + stride;
        unsigned idxn = 0;
        if (nxt < N) {
            idxn = hash3(coords[3 * nxt], coords[3 * nxt + 1], coords[3 * nxt + 2],
                         tmask, T);
            const float* pb = table + idxn;
#pragma unroll
            for (int c = 0; c < NF; ++c)
                __builtin_prefetch((const void*)(pb + (size_t)c * T), 0, 1);
        }

        // Gather current point's features (independent loads in flight).
        const float* base = table + idx;
        float f[NF];
#pragma unroll
        for (int c = 0; c < NF; ++c)
            f[c] = base[(size_t)c * T];

        if (CACHE) {
#pragma unroll
            for (int c = 0; c < NF; ++c)
                cache[(size_t)c * N + cur] = f[c];   // coalesced per row
        }

        s  += (double)f[0] + (double)f[1] + (double)f[2];
        ss += (double)f[0] * (double)f[0]
            + (double)f[1] * (double)f[1]
            + (double)f[2] * (double)f[2];

        idx = idxn;
    }

    // Deterministic block tree-reduction in LDS.
    __shared__ double sh[2 * BS];
    sh[threadIdx.x]      = s;
    sh[BS + threadIdx.x] = ss;
    __syncthreads();
    for (int o = BS / 2; o > 0; o >>= 1) {
        if ((int)threadIdx.x < o) {
            sh[threadIdx.x]      += sh[threadIdx.x + o];
            sh[BS + threadIdx.x] += sh[BS + threadIdx.x + o];
        }
        __syncthreads();
    }
    if (threadIdx.x == 0) {
        partials[2 * blockIdx.x]     = sh[0];
        partials[2 * blockIdx.x + 1] = sh[BS];
    }
}

// ---------------------------------------------------------------------------
// Pass 2 (single block): fold partials -> mean/std, precompute all scalar
// constants the finalize kernel needs.
//   scal[0]=mean  scal[1]=dm_scale  scal[2]=k(=2*far/vs)
//   scal[4..6]=cc[j]+far/vs-far (voxel-center offset)
// ---------------------------------------------------------------------------
__global__ void k_scalars(const double* __restrict__ partials, int nparts,
                          const float*  __restrict__ cc,
                          const float*  __restrict__ far,
                          const int*    __restrict__ vsz,
                          float* __restrict__ scal, int N) {
    __shared__ double sh[2 * BS];
    double s = 0.0, ss = 0.0;
    for (int i = threadIdx.x; i < nparts; i += BS) {
        s  += partials[2 * i];
        ss += partials[2 * i + 1];
    }
    sh[threadIdx.x]      = s;
    sh[BS + threadIdx.x] = ss;
    __syncthreads();
    for (int o = BS / 2; o > 0; o >>= 1) {
        if ((int)threadIdx.x < o) {
            sh[threadIdx.x]      += sh[threadIdx.x + o];
            sh[BS + threadIdx.x] += sh[BS + threadIdx.x + o];
        }
        __syncthreads();
    }
    if (threadIdx.x == 0) {
        double M    = 3.0 * (double)N;
        double sum  = sh[0];
        double sq   = sh[BS];
        double mean = sum / M;
        double var  = (sq - sum * sum / M) / (M - 1.0);  // ddof=1
        double stdv = sqrt(var);

        float far_s = far[0];
        float vs    = (float)vsz[0];
        float k     = 2.0f * far_s / vs;
        float off   = far_s / vs - far_s;

        scal[0] = (float)mean;
        scal[1] = (float)((double)k / 6.0 / stdv);  // dm scale factor
        scal[2] = k;
        scal[3] = 0.0f;
        scal[4] = cc[0] + off;
        scal[5] = cc[1] + off;
        scal[6] = cc[2] + off;
        scal[7] = 0.0f;
    }
}

// ---------------------------------------------------------------------------
// Pass 3: per-point finalize.
// CACHE path: TDM (tensor_load_to_lds) DMAs a 14 x 256 float tile of the
// feature cache into LDS (2D descriptor: row stride = N), overlapping with
// the coord/scalar loads; threads then read features from LDS
// (stride-256 -> bank-conflict free). Non-cache path re-gathers directly.
// ---------------------------------------------------------------------------
template <bool CACHE>
__global__ void k_final(const int*   __restrict__ coords,
                        const float* __restrict__ table,
                        const float* __restrict__ cache,
                        const float* __restrict__ scal,
                        float* __restrict__ out,
                        int N, unsigned T, unsigned tmask) {
    __shared__ float tile[14 * BS];

    const int i = blockIdx.x * BS + (int)threadIdx.x;
    const bool valid = (i < N);

    if (CACHE) {
        if (threadIdx.x == 0) {
            const int start = blockIdx.x * BS;
            unsigned long long ga =
                (unsigned long long)(const void*)(cache + (size_t)start);
            unsigned lds_off = (unsigned)(unsigned long long)(const void*)&tile[0];

            u32x4 g0;
            g0.x = 1u;                        // count=1, load, no gather
            g0.y = lds_off;                   // lds_addr (bytes)
            g0.z = (unsigned)ga;              // global_addr[31:0]
            g0.w = (unsigned)(ga >> 32) | 0x80000000u; // addr[56:32] | type=2

            unsigned nd = (unsigned)N;
            u32x8 g1;
            g1.s0 = 0x00020000u;              // data_size=2 (4 bytes)
            g1.s1 = (nd & 0xFFFFu) << 16;     // tensor_dim0[15:0]
            g1.s2 = (nd >> 16) | (14u << 16); // tensor_dim0[31:16] | tensor_dim1=14
            g1.s3 = ((unsigned)BS) << 16;     // tile_dim0 = 256
            g1.s4 = 14u;                      // tile_dim1 = 14, tile_dim2 = 0
            g1.s5 = nd;                       // tensor_dim0_stride[31:0] = N
            g1.s6 = 0u;                       // stride0[47:32] | stride1[15:0]
            g1.s7 = 0u;                       // stride1[47:16]

            asm volatile("tensor_load_to_lds %0, %1"
                         :: "s"(g0), "s"(g1) : "memory");
        }
    }

    // Overlap DMA with scalar-constant and coordinate loads.
    const float mean   = scal[0];
    const float dscale = scal[1];
    const float k      = scal[2];
    const float off0   = scal[4];
    const float off1   = scal[5];
    const float off2   = scal[6];

    int c0 = 0, c1 = 0, c2 = 0;
    if (valid) {
        c0 = coords[3 * i];
        c1 = coords[3 * i + 1];
        c2 = coords[3 * i + 2];
    }

    float f[14];
    if (CACHE) {
        if (threadIdx.x == 0)
            asm volatile("s_wait_tensorcnt 0x0" ::: "memory");
        __syncthreads();
#pragma unroll
        for (int c = 0; c < 14; ++c)
            f[c] = tile[c * BS + threadIdx.x];
    } else {
        unsigned idx = valid ? hash3(c0, c1, c2, tmask, T) : 0u;
        const float* base = table + idx;
#pragma unroll
        for (int c = 0; c < 14; ++c)
            f[c] = base[(size_t)c * T];
    }

    if (!valid) return;

    // means = (dm - mean)*dscale + voxel_center
    float m0 = (f[0] - mean) * dscale + (float)c0 * k + off0;
    float m1 = (f[1] - mean) * dscale + (float)c1 * k + off1;
    float m2 = (f[2] - mean) * dscale + (float)c2 * k + off2;

    // normalized quaternion -> rotation matrix
    float qr = f[3], qx = f[4], qy = f[5], qz = f[6];
    float inv = 1.0f / sqrtf(qr * qr + qx * qx + qy * qy + qz * qz);
    qr *= inv; qx *= inv; qy *= inv; qz *= inv;

    float s0 = sigmoidf(f[7]) * k;
    float s1 = sigmoidf(f[8]) * k;
    float s2 = sigmoidf(f[9]) * k;

    float R00 = 1.0f - 2.0f * (qy * qy + qz * qz);
    float R01 = 2.0f * (qx * qy - qr * qz);
    float R02 = 2.0f * (qx * qz + qr * qy);
    float R10 = 2.0f * (qx * qy + qr * qz);
    float R11 = 1.0f - 2.0f * (qx * qx + qz * qz);
    float R12 = 2.0f * (qy * qz - qr * qx);
    float R20 = 2.0f * (qx * qz - qr * qy);
    float R21 = 2.0f * (qy * qz + qr * qx);
    float R22 = 1.0f - 2.0f * (qx * qx + qy * qy);

    float L00 = R00 * s0, L01 = R01 * s1, L02 = R02 * s2;
    float L10 = R10 * s0, L11 = R11 * s1, L12 = R12 * s2;
    float L20 = R20 * s0, L21 = R21 * s1, L22 = R22 * s2;

    // cov = L * L^T (symmetric 3x3, 18 FMAs on VALU — cheaper than any
    // WMMA marshaling for per-point tiny matmuls)
    float C00 = L00 * L00 + L01 * L01 + L02 * L02;
    float C01 = L00 * L10 + L01 * L11 + L02 * L12;
    float C02 = L00 * L20 + L01 * L21 + L02 * L22;
    float C11 = L10 * L10 + L11 * L11 + L12 * L12;
    float C12 = L10 * L20 + L11 * L21 + L12 * L22;
    float C22 = L20 * L20 + L21 * L21 + L22 * L22;

    const size_t Ns = (size_t)N;
    // means (1,N,3)
    float* mp = out + 3 * (size_t)i;
    mp[0] = m0; mp[1] = m1; mp[2] = m2;
    // covariances (1,N,3,3)
    float* cp = out + 3 * Ns + 9 * (size_t)i;
    cp[0] = C00; cp[1] = C01; cp[2] = C02;
    cp[3] = C01; cp[4] = C11; cp[5] = C12;
    cp[6] = C02; cp[7] = C12; cp[8] = C22;
    // harmonics (1,N,3,1)
    float* hp = out + 12 * Ns + 3 * (size_t)i;
    hp[0] = f[10]; hp[1] = f[11]; hp[2] = f[12];
    // opacities (1,N)
    out[15 * Ns + (size_t)i] = sigmoidf(f[13] - 4.0f);
}

// ---------------------------------------------------------------------------
extern "C" void kernel_launch(void* const* d_in, const int* in_sizes, int n_in,
                              void* d_out, int out_size, void* d_ws, size_t ws_size,
                              hipStream_t stream) {
    const int*   coords = (const int*)d_in[0];
    const float* table  = (const float*)d_in[1];
    const float* cc     = (const float*)d_in[2];
    const float* far    = (const float*)d_in[3];
    const int*   vsz    = (const int*)d_in[4];

    const int N = in_sizes[0] / 3;
    const unsigned T = (unsigned)(in_sizes[1] / 14);
    const unsigned tmask = ((T & (T - 1)) == 0u && T != 0u) ? (T - 1u) : 0u;

    int grid_full = (N + BS - 1) / BS;
    int grid1 = grid_full < GRID1 ? grid_full : GRID1;

    // Workspace layout: [partials: grid1*2 doubles][scalars: 8 floats][feature cache]
    size_t scal_off = ((size_t)grid1 * 2 * sizeof(double) + 255) & ~(size_t)255;
    size_t feat_off = (scal_off + 256 + 255) & ~(size_t)255;
    // +4096 slack: last TDM tile may over-read up to ~1KB past the cache rows.
    size_t need_cache = feat_off + (size_t)14 * (size_t)N * sizeof(float) + 4096;
    bool useCache = ws_size >= need_cache;

    double* partials = (double*)d_ws;
    float*  scal     = (float*)((char*)d_ws + scal_off);
    float*  cache    = (float*)((char*)d_ws + feat_off);

    if (useCache)
        k_gather_reduce<true><<<grid1, BS, 0, stream>>>(coords, table, cache,
                                                        partials, N, T, tmask);
    else
        k_gather_reduce<false><<<grid1, BS, 0, stream>>>(coords, table, nullptr,
                                                         partials, N, T, tmask);

    k_scalars<<<1, BS, 0, stream>>>(partials, grid1, cc, far, vsz, scal, N);

    if (useCache)
        k_final<true><<<grid_full, BS, 0, stream>>>(coords, table, cache, scal,
                                                    (float*)d_out, N, T, tmask);
    else
        k_final<false><<<grid_full, BS, 0, stream>>>(coords, table, nullptr, scal,
                                                     (float*)d_out, N, T, tmask);
}